// GlobalStreamEncoder_68247030333977
// MI455X (gfx1250) — compile-verified
//
#include <hip/hip_runtime.h>
#include <hip/hip_bf16.h>
#include <stdint.h>

// ---------------------------------------------------------------------------
// GNN forward for MI455X (gfx1250, wave32).
//  - adjacency as 32MB bit-matrix (L2-resident), exact duplicate-collapse
//  - neighbor aggregation: deterministic ascending-j bit scan, coalesced h rows
//  - all dense GEMMs via V_WMMA_F32_16X16X4_F32 (full f32 precision),
//    A tile + pair-interleaved W panel staged in LDS -> K-loop is
//    ds_load_b64 + v_wmma only (no register packing moves)
// ---------------------------------------------------------------------------

#define NNODES 16384
#define NEDGES 524288
#define HID    64
#define WORDS_PER_ROW 512   // 16384 bits / 32

typedef __attribute__((ext_vector_type(2))) float v2f;
typedef __attribute__((ext_vector_type(8))) float v8f;

// ---------------- adjacency bitmask ----------------------------------------

__global__ void zero_u32_kernel(uint32_t* __restrict__ p, int n) {
    int i = blockIdx.x * blockDim.x + threadIdx.x;
    int stride = gridDim.x * blockDim.x;
    for (; i < n; i += stride) p[i] = 0u;
}

__global__ void build_adj_kernel(const long long* __restrict__ ei,
                                 uint32_t* __restrict__ bits, int nE) {
    int e = blockIdx.x * blockDim.x + threadIdx.x;
    if (e >= nE) return;
    int s = (int)ei[e];        // edge_index[0][e]
    int d = (int)ei[nE + e];   // edge_index[1][e]
    atomicOr(&bits[(size_t)s * WORDS_PER_ROW + (d >> 5)], 1u << (d & 31));
}

// neighbor_sum[i][c] = sum over set bits j of row i of h[j][c]
// one block (64 threads) per node row; thread t owns channel t.
// Word loads are wave-uniform (broadcast); h-row gathers are coalesced 256B.
__global__ __launch_bounds__(HID) void nsum_kernel(
    const uint32_t* __restrict__ bits, const float* __restrict__ h,
    float* __restrict__ out) {
    const int row = blockIdx.x;
    const int t = threadIdx.x;
    const uint32_t* rowbits = bits + (size_t)row * WORDS_PER_ROW;
    float acc = 0.f;
    for (int wi = 0; wi < WORDS_PER_ROW; ++wi) {
        uint32_t w = rowbits[wi];
        while (w) {
            int b = __builtin_ctz(w);
            w &= (w - 1u);
            int j = (wi << 5) + b;
            acc += h[(size_t)j * HID + t];
        }
    }
    out[(size_t)row * HID + t] = acc;
}

// ---------------- generic WMMA f32 GEMM ------------------------------------
// C[M,64] = concat(A1[:,0:K1], A2[:,0:K2], zeros) @ W[K1+K2,64] + bias
// Kpad = K1+K2 rounded up to a multiple of 4.
// 128 threads = 4 waves per block; block covers 64 rows, wave -> 16x64 tile.
// LDS staging:
//   Atile[64][Kpad]                        (row-major, 8B-aligned v2f reads)
//   Wlds pair-interleaved: element (k,n) at [(k>>1)*128 + n*2 + (k&1)]
//     -> B fragment {W[ka][n], W[ka+1][n]} (ka even) is ONE ds_load_b64
//        straight into an even-aligned VGPR pair.

__global__ __launch_bounds__(128) void wmma_gemm_kernel(
    const float* __restrict__ A1, int K1,
    const float* __restrict__ A2, int K2,
    const float* __restrict__ W, const float* __restrict__ bias,
    float* __restrict__ Cout, int Kpad) {
    __shared__ float Wlds[128 * 64];    // up to K=128 panel: 32 KB
    __shared__ float Atile[64 * 128];   // 64 rows x Kpad     : 32 KB

    const int Ktot = K1 + K2;
    const int mBase = blockIdx.x * 64;

    // W panel, pair-interleaved (rows >= Ktot zero-padded)
    for (int i = threadIdx.x; i < Kpad * 64; i += 128) {
        int k = i >> 6;
        int n = i & 63;
        float v = (k < Ktot) ? W[i] : 0.f;
        Wlds[((k >> 1) << 7) + (n << 1) + (k & 1)] = v;
    }
    // A1 region (coalesced along k within each row)
    for (int i = threadIdx.x; i < 64 * K1; i += 128) {
        int r = i / K1, k = i - r * K1;
        Atile[r * Kpad + k] = A1[(size_t)(mBase + r) * K1 + k];
    }
    // A2 region
    if (K2 > 0) {
        for (int i = threadIdx.x; i < 64 * K2; i += 128) {
            int r = i / K2, k = i - r * K2;
            Atile[r * Kpad + K1 + k] = A2[(size_t)(mBase + r) * K2 + k];
        }
    }
    // zero padding region (Kpad > Ktot only for the in-projection, 6 -> 8)
    {
        const int pad = Kpad - Ktot;
        for (int i = threadIdx.x; i < 64 * pad; i += 128) {
            int r = i / pad, k = i - r * pad;
            Atile[r * Kpad + Ktot + k] = 0.f;
        }
    }
    __syncthreads();

    const int wave = threadIdx.x >> 5;   // wave32 (gfx1250)
    const int lane = threadIdx.x & 31;
    const int hh = lane >> 4;            // lane half selects K pair
    const int ln = lane & 15;
    const int m0 = mBase + wave * 16;

    v8f acc[4] = {};                     // 4 N-tiles of 16 -> N=64
    const float* Arow = &Atile[(wave * 16 + ln) * Kpad];

    for (int k0 = 0; k0 < Kpad; k0 += 4) {
        // A 16x4 f32 fragment: vgpr v holds A[m][k0 + 2*hh + v]
        const int ka = k0 + 2 * hh;       // always even
        const v2f a = *(const v2f*)(Arow + ka);            // ds_load_b64
        const float* Wp = &Wlds[(ka >> 1) << 7];
#pragma unroll
        for (int tI = 0; tI < 4; ++tI) {
            // B 4x16 f32 fragment: vgpr v holds W[ka + v][n], contiguous pair
            const int nn = tI * 16 + ln;
            const v2f b = *(const v2f*)(Wp + (nn << 1));   // ds_load_b64
            acc[tI] = __builtin_amdgcn_wmma_f32_16x16x4_f32(
                false, a, false, b, (short)0, acc[tI], false, false);
        }
    }

    // C/D layout: vgpr r -> M = r (lanes 0-15) / r+8 (lanes 16-31); N = ln
#pragma unroll
    for (int tI = 0; tI < 4; ++tI) {
        const int col = tI * 16 + ln;
        const float bv = bias[col];
#pragma unroll
        for (int r = 0; r < 8; ++r) {
            const int row = m0 + r + hh * 8;
            Cout[(size_t)row * 64 + col] = acc[tI][r] + bv;
        }
    }
}

// ---------------- LayerNorm + ReLU (+ residual) ----------------------------
// one block (64 threads) per row, C=64 channels.

__global__ __launch_bounds__(HID) void ln_relu_kernel(
    const float* __restrict__ z, const float* __restrict__ g,
    const float* __restrict__ b, const float* __restrict__ resid,
    float* __restrict__ out) {
    const int row = blockIdx.x;
    const int t = threadIdx.x;
    __shared__ float red[HID];

    float x = z[(size_t)row * HID + t];

    red[t] = x;
    __syncthreads();
#pragma unroll
    for (int s = HID / 2; s > 0; s >>= 1) {
        if (t < s) red[t] += red[t + s];
        __syncthreads();
    }
    float mean = red[0] * (1.f / HID);
    __syncthreads();

    float d = x - mean;
    red[t] = d * d;
    __syncthreads();
#pragma unroll
    for (int s = HID / 2; s > 0; s >>= 1) {
        if (t < s) red[t] += red[t + s];
        __syncthreads();
    }
    float var = red[0] * (1.f / HID);

    float y = d * rsqrtf(var + 1e-5f) * g[t] + b[t];
    y = fmaxf(y, 0.f);
    if (resid) y += resid[(size_t)row * HID + t];
    out[(size_t)row * HID + t] = y;
}

// ---------------------------------------------------------------------------

extern "C" void kernel_launch(void* const* d_in, const int* in_sizes, int n_in,
                              void* d_out, int out_size, void* d_ws, size_t ws_size,
                              hipStream_t stream) {
    // setup_inputs() order
    const float*     node_features = (const float*)d_in[0];       // [N,6]
    const long long* edge_index    = (const long long*)d_in[1];   // [2,E] int64
    const float*     tpe   = (const float*)d_in[2];               // [N,8]
    const float*     in_W  = (const float*)d_in[3];               // [6,64]
    const float*     in_b  = (const float*)d_in[4];               // [64]
    const float*     layer_W = (const float*)d_in[5];             // [3,128,64]
    const float*     layer_b = (const float*)d_in[6];             // [3,64]
    const float*     ln_g  = (const float*)d_in[7];               // [3,64]
    const float*     ln_b  = (const float*)d_in[8];               // [3,64]
    const float*     out_W = (const float*)d_in[9];               // [64,64]
    const float*     out_b = (const float*)d_in[10];              // [64]
    const float*     tpe_W = (const float*)d_in[11];              // [8,64]
    const float*     tpe_b = (const float*)d_in[12];              // [64]
    const float*     proj_W = (const float*)d_in[13];             // [128,64]
    const float*     proj_b = (const float*)d_in[14];             // [64]
    const float*     pln_g = (const float*)d_in[15];              // [64]
    const float*     pln_b = (const float*)d_in[16];              // [64]
    (void)in_sizes; (void)n_in; (void)out_size; (void)ws_size;

    // workspace layout
    char* ws = (char*)d_ws;
    const size_t bits_bytes = (size_t)NNODES * WORDS_PER_ROW * 4;   // 32 MB
    const size_t mat_bytes  = (size_t)NNODES * HID * 4;             // 4 MB
    uint32_t* bits = (uint32_t*)ws;                ws += bits_bytes;
    float* h_a  = (float*)ws;                      ws += mat_bytes;
    float* h_b  = (float*)ws;                      ws += mat_bytes;
    float* nsum = (float*)ws;                      ws += mat_bytes;
    float* zbuf = (float*)ws;                      ws += mat_bytes;
    float* gbuf = (float*)ws;                      ws += mat_bytes;
    float* tbuf = (float*)ws;                      ws += mat_bytes;

    const int nWords = NNODES * WORDS_PER_ROW;
    const dim3 gemmGrid(NNODES / 64);   // 4 waves * 16 rows per block
    const dim3 gemmBlk(128);

    // 1) adjacency bitmask
    zero_u32_kernel<<<4096, 256, 0, stream>>>(bits, nWords);
    build_adj_kernel<<<(NEDGES + 255) / 256, 256, 0, stream>>>(edge_index, bits, NEDGES);

    // 2) h = X @ in_W + in_b   (K1=6 padded to 8)
    wmma_gemm_kernel<<<gemmGrid, gemmBlk, 0, stream>>>(
        node_features, 6, nullptr, 0, in_W, in_b, h_a, 8);

    // 3) GNN layers
    float* hcur = h_a;
    float* hnxt = h_b;
    for (int l = 0; l < 3; ++l) {
        nsum_kernel<<<NNODES, HID, 0, stream>>>(bits, hcur, nsum);
        wmma_gemm_kernel<<<gemmGrid, gemmBlk, 0, stream>>>(
            hcur, 64, nsum, 64, layer_W + (size_t)l * 128 * 64,
            layer_b + (size_t)l * 64, zbuf, 128);
        ln_relu_kernel<<<NNODES, HID, 0, stream>>>(
            zbuf, ln_g + (size_t)l * 64, ln_b + (size_t)l * 64, hcur, hnxt);
        float* tmp = hcur; hcur = hnxt; hnxt = tmp;
    }

    // 4) g = h @ out_W + out_b
    wmma_gemm_kernel<<<gemmGrid, gemmBlk, 0, stream>>>(
        hcur, 64, nullptr, 0, out_W, out_b, gbuf, 64);

    // 5) t = tpe @ tpe_W + tpe_b
    wmma_gemm_kernel<<<gemmGrid, gemmBlk, 0, stream>>>(
        tpe, 8, nullptr, 0, tpe_W, tpe_b, tbuf, 8);

    // 6) z = [g | t] @ proj_W + proj_b ; out = relu(LN(z))
    wmma_gemm_kernel<<<gemmGrid, gemmBlk, 0, stream>>>(
        gbuf, 64, tbuf, 64, proj_W, proj_b, zbuf, 128);
    ln_relu_kernel<<<NNODES, HID, 0, stream>>>(
        zbuf, pln_g, pln_b, nullptr, (float*)d_out);
}